// Net_44693429682822
// MI455X (gfx1250) — compile-verified
//
#include <hip/hip_runtime.h>
#include <hip/hip_bf16.h>

// ---------------------------------------------------------------------------
// GIN (3x GINConv + global-add-pool + MLP head) for gfx1250 / MI455X.
//   N=50000 nodes, E=600000 edges, D=128, G=128 graphs, OUT=10, L=3
// Strategy:
//   * scatter-add aggregation with global_atomic_add_f32 (wave-per-edge)
//   * node GEMMs via V_WMMA_F32_16X16X4_F32 (exact fp32)
//       - 64-row blocks, A staged in padded LDS (ds_load_2addr_b64 feeds A)
//       - one B register pair reused across 4 row tiles (4x fewer W loads)
//   * fused bias + BatchNorm(eval) + ReLU epilogue
// ---------------------------------------------------------------------------

#define D_FEAT   128
#define N_NODES  50000
#define N_EDGES  600000
#define N_GRAPHS 128
#define N_LAYERS 3
#define N_OUT    10
#define BN_INV   0.9999950000374997f   // 1/sqrt(1 + 1e-5)

#define ROWS_PER_BLOCK 64
#define LDS_PITCH 132                  // 128 + 4 pad: breaks stride-128 bank conflicts

typedef __attribute__((ext_vector_type(2))) float v2f;
typedef __attribute__((ext_vector_type(8))) float v8f;

// ---------------------------------------------------------------- copy h->agg
__global__ __launch_bounds__(256) void copy_f4(const float4* __restrict__ src,
                                               float4* __restrict__ dst, int n4) {
  int i = blockIdx.x * 256 + threadIdx.x;
  if (i < n4) dst[i] = src[i];
}

// ------------------------------------------------- edge scatter: agg[dst]+=x[src]
// one wave (32 lanes) per edge; each lane moves float4 (32*4 = 128 features)
__global__ __launch_bounds__(256) void scatter_edges(
    const int* __restrict__ srcIdx, const int* __restrict__ dstIdx,
    const float* __restrict__ x, float* __restrict__ agg, int nEdges) {
  int gid  = blockIdx.x * 256 + threadIdx.x;
  int edge = gid >> 5;
  int lane = gid & 31;
  if (edge >= nEdges) return;
  int s = srcIdx[edge];
  int d = dstIdx[edge];
  float4 v = ((const float4*)(x + (size_t)s * D_FEAT))[lane];
  float* dp = agg + (size_t)d * D_FEAT + lane * 4;
  atomicAdd(dp + 0, v.x);
  atomicAdd(dp + 1, v.y);
  atomicAdd(dp + 2, v.z);
  atomicAdd(dp + 3, v.w);
}

// ------------------------------------------------------------- WMMA GEMM 128-K
// out[M x 128] = relu((A[M x 128] @ W[128 x 128] + bias) * scale + shift)
// scale==nullptr -> scale 1 / shift 0 (plain linear + relu).
// Block: 256 thr = 8 waves; block covers 64 rows, wave w covers cols [16w,16w+16).
// Each wave: 4 row tiles x 1 col tile, sharing one B fragment per k-step.
__global__ __launch_bounds__(256) void gemm128_bn_relu_wmma(
    const float* __restrict__ A, const float* __restrict__ W,
    const float* __restrict__ bias, const float* __restrict__ scale,
    const float* __restrict__ shift, float* __restrict__ out, int M) {
  __shared__ float As[ROWS_PER_BLOCK * LDS_PITCH];   // 64 x 128 fp32, padded

  const int row0 = blockIdx.x * ROWS_PER_BLOCK;
  const int tid  = threadIdx.x;

  // stage 64x128 A tile -> LDS (2048 float4, 8 per thread); zero-fill OOB rows
  {
#pragma unroll
    for (int it = 0; it < 8; ++it) {
      int j  = tid + it * 256;          // float4 index within tile
      int r  = j >> 5;                  // row 0..63
      int c4 = j & 31;                  // float4 col 0..31
      int gr = row0 + r;
      float4 v;
      if (gr < M) v = *(const float4*)(A + (size_t)gr * D_FEAT + c4 * 4);
      else        v = make_float4(0.f, 0.f, 0.f, 0.f);
      *(float4*)&As[r * LDS_PITCH + c4 * 4] = v;
    }
  }
  __syncthreads();

  const int wave = tid >> 5;
  const int lane = tid & 31;
  const int col0 = wave * 16;
  const int n    = lane & 15;           // out column / A row within tile
  const int half = lane >> 4;           // 0: K {0,1}, rows 0-7 ; 1: K {2,3}, rows 8-15
  const int kq   = half * 2;

  v8f acc[4] = {{}, {}, {}, {}};
#pragma unroll 4
  for (int k = 0; k < D_FEAT; k += 4) {
    v2f b;                               // W[k+kq .. k+kq+1][col0+n], shared by 4 tiles
    b.x = W[(size_t)(k + kq)     * D_FEAT + col0 + n];
    b.y = W[(size_t)(k + kq + 1) * D_FEAT + col0 + n];
#pragma unroll
    for (int t = 0; t < 4; ++t) {
      const float* ar = &As[(t * 16 + n) * LDS_PITCH + k + kq];
      v2f a; a.x = ar[0]; a.y = ar[1];
      acc[t] = __builtin_amdgcn_wmma_f32_16x16x4_f32(
          /*neg_a=*/false, a, /*neg_b=*/false, b,
          /*c_mod=*/(short)0, acc[t], /*reuse_a=*/false, /*reuse_b=*/false);
    }
  }

  // fused epilogue: (acc + bias) * (g * 1/sqrt(1+eps)) + beta, then ReLU
  const int c  = col0 + n;
  const float bi = bias[c];
  const float sc = scale ? scale[c] * BN_INV : 1.0f;
  const float sh = shift ? shift[c] : 0.0f;
#pragma unroll
  for (int t = 0; t < 4; ++t) {
#pragma unroll
    for (int r = 0; r < 8; ++r) {
      int gr = row0 + t * 16 + r + half * 8;   // C/D layout: VGPR r -> row r (+8 hi half)
      if (gr < M) {
        float v = (acc[t][r] + bi) * sc + sh;
        out[(size_t)gr * D_FEAT + c] = fmaxf(v, 0.0f);
      }
    }
  }
}

// ------------------------------------------------------------------- pooling
__global__ __launch_bounds__(256) void zero_f(float* __restrict__ p, int n) {
  int i = blockIdx.x * 256 + threadIdx.x;
  if (i < n) p[i] = 0.0f;
}

__global__ __launch_bounds__(256) void pool_scatter(
    const float* __restrict__ h, const int* __restrict__ batch,
    float* __restrict__ pooled, int nElems) {
  int i = blockIdx.x * 256 + threadIdx.x;
  if (i >= nElems) return;
  int node = i >> 7;          // / 128
  int dim  = i & 127;
  atomicAdd(pooled + (size_t)batch[node] * D_FEAT + dim, h[i]);
}

// ------------------------------------------------------ final 128x10 projection
__global__ __launch_bounds__(32) void final_linear(
    const float* __restrict__ Hin, const float* __restrict__ W2,
    const float* __restrict__ b2, float* __restrict__ out) {
  int g = blockIdx.x;
  int o = threadIdx.x;
  if (o >= N_OUT) return;
  float s = b2[o];
#pragma unroll 8
  for (int k = 0; k < D_FEAT; ++k)
    s += Hin[(size_t)g * D_FEAT + k] * W2[(size_t)k * N_OUT + o];
  out[(size_t)g * N_OUT + o] = s;
}

// ---------------------------------------------------------------------------
extern "C" void kernel_launch(void* const* d_in, const int* in_sizes, int n_in,
                              void* d_out, int out_size, void* d_ws, size_t ws_size,
                              hipStream_t stream) {
  const float* x       = (const float*)d_in[0];
  const int*   eSrc    = (const int*)d_in[1];               // edge_index[0]
  const int*   eDst    = ((const int*)d_in[1]) + N_EDGES;   // edge_index[1]
  const int*   batch   = (const int*)d_in[2];
  const float* convW1  = (const float*)d_in[3];   // [L,D,D]
  const float* convb1  = (const float*)d_in[4];   // [L,D]
  const float* convg   = (const float*)d_in[5];   // [L,D]
  const float* convbt  = (const float*)d_in[6];   // [L,D]
  const float* convW2  = (const float*)d_in[7];   // [L,D,D]
  const float* convb2  = (const float*)d_in[8];   // [L,D]
  const float* mW1     = (const float*)d_in[9];
  const float* mb1     = (const float*)d_in[10];
  const float* mg      = (const float*)d_in[11];
  const float* mbt     = (const float*)d_in[12];
  const float* mW2     = (const float*)d_in[13];
  const float* mb2     = (const float*)d_in[14];

  const size_t ND = (size_t)N_NODES * D_FEAT;   // 6.4M floats
  float* agg     = (float*)d_ws;                // [N,D]
  float* tmp     = agg + ND;                    // [N,D]
  float* h       = tmp + ND;                    // [N,D]
  float* pooled  = h + ND;                      // [G,D]
  float* pooled2 = tmp;                         // reuse tmp for head hidden

  const int cpBlocks   = (int)(ND / 4 + 255) / 256;                 // float4 copy
  const int scBlocks   = (N_EDGES * 32 + 255) / 256;                // wave per edge
  const int gemmBlocks = (N_NODES + ROWS_PER_BLOCK - 1) / ROWS_PER_BLOCK;  // 782
  const int poolBlocks = (int)(ND + 255) / 256;

  const float* hcur = x;
  for (int l = 0; l < N_LAYERS; ++l) {
    // agg = h ; agg[dst] += h[src]
    copy_f4<<<cpBlocks, 256, 0, stream>>>((const float4*)hcur, (float4*)agg,
                                          (int)(ND / 4));
    scatter_edges<<<scBlocks, 256, 0, stream>>>(eSrc, eDst, hcur, agg, N_EDGES);
    // tmp = relu(BN(agg @ W1 + b1))
    gemm128_bn_relu_wmma<<<gemmBlocks, 256, 0, stream>>>(
        agg, convW1 + (size_t)l * D_FEAT * D_FEAT, convb1 + l * D_FEAT,
        convg + l * D_FEAT, convbt + l * D_FEAT, tmp, N_NODES);
    // h = relu(tmp @ W2 + b2)   (outer .relu() fused here)
    gemm128_bn_relu_wmma<<<gemmBlocks, 256, 0, stream>>>(
        tmp, convW2 + (size_t)l * D_FEAT * D_FEAT, convb2 + l * D_FEAT,
        nullptr, nullptr, h, N_NODES);
    hcur = h;
  }

  // pooled[g] = sum over nodes in graph g
  zero_f<<<(N_GRAPHS * D_FEAT + 255) / 256, 256, 0, stream>>>(pooled,
                                                             N_GRAPHS * D_FEAT);
  pool_scatter<<<poolBlocks, 256, 0, stream>>>(h, batch, pooled, (int)ND);

  // head: pooled2 = relu(BN(pooled @ mW1 + mb1)) ; out = pooled2 @ mW2 + mb2
  gemm128_bn_relu_wmma<<<(N_GRAPHS + ROWS_PER_BLOCK - 1) / ROWS_PER_BLOCK, 256, 0,
                        stream>>>(pooled, mW1, mb1, mg, mbt, pooled2, N_GRAPHS);
  final_linear<<<N_GRAPHS, 32, 0, stream>>>(pooled2, mW2, mb2, (float*)d_out);
}